// VectorQuantizer_29111288332979
// MI455X (gfx1250) — compile-verified
//
#include <hip/hip_runtime.h>

// CDNA5 / gfx1250 VQ-VAE vector quantizer.
// fp32 WMMA (v_wmma_f32_16x16x4_f32) for the [N,64] x [64,1024] distance matmul.
// Each wave32 owns TWO 16-row tiles: one set of codebook B-fragments feeds two
// independent WMMA accumulator chains (ILP=2, halves codebook L2 traffic).

typedef __attribute__((ext_vector_type(2))) float v2f;
typedef __attribute__((ext_vector_type(8))) float v8f;

#define K_ENTRIES 1024
#define C_DIM     64
#define N_ROWS    65536             // 64 * 32 * 32
#define WAVES_TOTAL (N_ROWS / 32)   // 2048 waves, 32 rows (2 tiles) each
#define WAVES_PER_BLOCK 4
#define OUT_ELEMS 4194304           // 64*64*32*32

// ---------------------------------------------------------------------------
// Kernel 1: precompute ||e||^2 per codebook entry into d_ws[0..1023].
// ---------------------------------------------------------------------------
__global__ __launch_bounds__(256) void vq_init_e2(const float* __restrict__ emb,
                                                  float* __restrict__ e2) {
  int i = blockIdx.x * blockDim.x + threadIdx.x;
  if (i < K_ENTRIES) {
    const float* row = emb + i * C_DIM;
    float s = 0.0f;
#pragma unroll
    for (int c = 0; c < C_DIM; ++c) { float v = row[c]; s += v * v; }
    e2[i] = s;
  }
}

// ---------------------------------------------------------------------------
// Kernel 2: main quantizer. One wave32 per pair of 16-row tiles.
//
// Layouts (CDNA5 ISA 7.12.2, 32-bit 16x16x4 WMMA):
//   A (16x4, MxK): lane L -> row M = L&15; VGPR0/1 hold K = 2*(L>>4) + {0,1}
//   B (4x16, KxN): lane L -> col N = L&15; VGPR0/1 hold K = 2*(L>>4) + {0,1}
//   D (16x16):     lane L -> col N = L&15; VGPR j holds row M = j + 8*(L>>4)
//
// latents are [B=64, C=64, H=32, W=32]; flat row n = b*1024 + hw, element
// (n, c) lives at lat[b*65536 + c*1024 + hw].  Wave base n0 is a multiple of
// 32 and 1024 % 32 == 0, so both tiles stay inside one batch image.
// ---------------------------------------------------------------------------
__global__ __launch_bounds__(32 * WAVES_PER_BLOCK) void vq_main(
    const float* __restrict__ lat, const float* __restrict__ emb,
    const float* __restrict__ e2, float* __restrict__ out,
    float* __restrict__ partials) {
  const int lane = threadIdx.x & 31;
  const int wave = threadIdx.x >> 5;
  const int w = blockIdx.x * WAVES_PER_BLOCK + wave;  // 0..2047
  const int n0 = w * 32;
  const int b = n0 >> 10;
  const int hw0 = n0 & 1023;
  const float* latb = lat + b * (C_DIM * 1024) + hw0;

  const int r = lane & 15;            // row within tile (A) / entry-in-chunk (B)
  const int ch_off = (lane >> 4) * 2; // K sub-offset within each 4-wide step

  // ---- load A fragments once for both tiles; reused for all 64 chunks.
  v2f a0[16], a1[16];
#pragma unroll
  for (int s = 0; s < 16; ++s) {
    const int c = 4 * s + ch_off;
    a0[s].x = latb[c * 1024 + r];
    a0[s].y = latb[(c + 1) * 1024 + r];
    a1[s].x = latb[c * 1024 + 16 + r];
    a1[s].y = latb[(c + 1) * 1024 + 16 + r];
  }

  float mv0[8], mv1[8];
  int mi0[8], mi1[8];
#pragma unroll
  for (int j = 0; j < 8; ++j) {
    mv0[j] = 3.4e38f; mi0[j] = 0;
    mv1[j] = 3.4e38f; mi1[j] = 0;
  }

  // ---- sweep the codebook, 16 entries per chunk, K accumulated via WMMA.
  for (int chunk = 0; chunk < 64; ++chunk) {
    const int entry = chunk * 16 + r;            // this lane's entry (N = r)
    const float* embrow = emb + entry * C_DIM + ch_off;

    // Load all 16 B fragments up front (one clause of global_load_b64).
    v2f bf[16];
#pragma unroll
    for (int s = 0; s < 16; ++s) {
      bf[s].x = embrow[4 * s];
      bf[s].y = embrow[4 * s + 1];
    }

    // Two independent accumulator chains share the B fragments.
    v8f acc0 = {}, acc1 = {};
#pragma unroll
    for (int s = 0; s < 16; ++s) {
      acc0 = __builtin_amdgcn_wmma_f32_16x16x4_f32(
          false, a0[s], false, bf[s], (short)0, acc0, false, false);
      acc1 = __builtin_amdgcn_wmma_f32_16x16x4_f32(
          false, a1[s], false, bf[s], (short)0, acc1, false, false);
    }

    const float se2 = e2[entry];
#pragma unroll
    for (int j = 0; j < 8; ++j) {
      // score = ||e||^2 - 2 x.e  (||x||^2 is constant per row -> dropped)
      const float c0 = se2 - 2.0f * acc0[j];
      if (c0 < mv0[j]) { mv0[j] = c0; mi0[j] = entry; }
      const float c1 = se2 - 2.0f * acc1[j];
      if (c1 < mv1[j]) { mv1[j] = c1; mi1[j] = entry; }
    }
  }

  // ---- argmin across the 16 lanes that share each row (xor stays in-half).
#pragma unroll
  for (int j = 0; j < 8; ++j) {
#pragma unroll
    for (int m = 1; m < 16; m <<= 1) {
      float ov = __shfl_xor(mv0[j], m, 32);
      int oi = __shfl_xor(mi0[j], m, 32);
      if (ov < mv0[j] || (ov == mv0[j] && oi < mi0[j])) { mv0[j] = ov; mi0[j] = oi; }
      ov = __shfl_xor(mv1[j], m, 32);
      oi = __shfl_xor(mi1[j], m, 32);
      if (ov < mv1[j] || (ov == mv1[j] && oi < mi1[j])) { mv1[j] = ov; mi1[j] = oi; }
    }
  }

  // ---- broadcast winning indices through LDS.
  // Rows 0-7 live in the low-half lanes' slots, rows 8-15 in the high half.
  __shared__ int widx[WAVES_PER_BLOCK][32];
  if ((lane & 15) == 0) {
    const int base = (lane >> 4) * 8;
#pragma unroll
    for (int j = 0; j < 8; ++j) {
      widx[wave][base + j] = mi0[j];
      widx[wave][16 + base + j] = mi1[j];
    }
  }
  __syncthreads();

  // ---- write quantized output (coalesced over hw) + accumulate loss.
  float lacc = 0.0f;
  float* outb = out + b * (C_DIM * 1024) + hw0;
#pragma unroll
  for (int i = 0; i < 32; ++i) {
    const int c = 2 * i + (lane >> 4);
    const int rr = lane & 15;
    const int coff = c * 1024 + rr;
    // tile 0 (rows 0-15)
    const int e0i = widx[wave][rr];
    const float q0 = emb[e0i * C_DIM + c];
    const float l0 = latb[coff];
    outb[coff] = q0;
    const float d0 = l0 - q0;
    lacc += d0 * d0;
    // tile 1 (rows 16-31)
    const int e1i = widx[wave][16 + rr];
    const float q1 = emb[e1i * C_DIM + c];
    const float l1 = latb[coff + 16];
    outb[coff + 16] = q1;
    const float d1 = l1 - q1;
    lacc += d1 * d1;
  }
#pragma unroll
  for (int m = 1; m < 32; m <<= 1) lacc += __shfl_xor(lacc, m, 32);
  if (lane == 0) partials[w] = lacc;  // deterministic: no float atomics
}

// ---------------------------------------------------------------------------
// Kernel 3: fixed-order loss reduction -> d_out[OUT_ELEMS].
// vq_loss = (1 + 0.25) * mean((lat - quant)^2)
// ---------------------------------------------------------------------------
__global__ __launch_bounds__(256) void vq_finalize(const float* __restrict__ partials,
                                                   float* __restrict__ out_scalar) {
  __shared__ float red[256];
  float s = 0.0f;
  for (int t = threadIdx.x; t < WAVES_TOTAL; t += 256) s += partials[t];
  red[threadIdx.x] = s;
  __syncthreads();
  for (int off = 128; off > 0; off >>= 1) {
    if ((int)threadIdx.x < off) red[threadIdx.x] += red[threadIdx.x + off];
    __syncthreads();
  }
  if (threadIdx.x == 0) *out_scalar = 1.25f * red[0] / (float)OUT_ELEMS;
}

extern "C" void kernel_launch(void* const* d_in, const int* in_sizes, int n_in,
                              void* d_out, int out_size, void* d_ws, size_t ws_size,
                              hipStream_t stream) {
  (void)in_sizes; (void)n_in; (void)out_size; (void)ws_size;
  const float* lat = (const float*)d_in[0];   // [64, 64, 32, 32] f32
  const float* emb = (const float*)d_in[1];   // [1024, 64] f32
  float* out = (float*)d_out;                 // [64,64,32,32] f32 ++ [1] f32

  float* e2 = (float*)d_ws;                   // 1024 floats
  float* partials = e2 + K_ENTRIES;           // 2048 floats

  vq_init_e2<<<(K_ENTRIES + 255) / 256, 256, 0, stream>>>(emb, e2);
  vq_main<<<WAVES_TOTAL / WAVES_PER_BLOCK, 32 * WAVES_PER_BLOCK, 0, stream>>>(
      lat, emb, e2, out, partials);
  vq_finalize<<<1, 256, 0, stream>>>(partials, out + OUT_ELEMS);
}